// EdgeEnhancedGATConv_2697239462579
// MI455X (gfx1250) — compile-verified
//
#include <hip/hip_runtime.h>
#include <hip/hip_bf16.h>

#define N_NODES   40000
#define N_EDGES   640000
#define IN_FEATS  256
#define EDGE_FEATS 64
#define OUT_FEATS 128
#define NUM_HEADS 8
// D = 16

typedef __attribute__((ext_vector_type(2))) float v2f;
typedef __attribute__((ext_vector_type(8))) float v8f;

// -------- float atomic max via integer atomics (works with -inf init) --------
__device__ __forceinline__ void atomicMaxFloat(float* addr, float val) {
    if (val >= 0.0f) {
        atomicMax((int*)addr, __float_as_int(val));
    } else {
        atomicMin((unsigned int*)addr, __float_as_uint(val));
    }
}

// -------- 0) init: out = 0, m = -inf, z = 0 --------
__global__ __launch_bounds__(256) void init_kernel(float* __restrict__ out,
                                                   float* __restrict__ mmax,
                                                   float* __restrict__ z) {
    int gid = blockIdx.x * blockDim.x + threadIdx.x;
    if (gid < N_NODES * OUT_FEATS) out[gid] = 0.0f;
    if (gid < N_NODES * NUM_HEADS) {
        mmax[gid] = -__builtin_inff();
        z[gid] = 0.0f;
    }
}

// -------- 1) fold a_e into edge_w: wcomb[h][k] = sum_d a_e[h,d]*edge_w[h*16+d][k] --------
__global__ __launch_bounds__(512) void prep_wcomb(const float* __restrict__ edge_w,
                                                  const float* __restrict__ edge_b,
                                                  const float* __restrict__ attn,
                                                  float* __restrict__ wcomb,
                                                  float* __restrict__ bcomb) {
    int tid = threadIdx.x;            // 512 threads = 8 heads * 64 k
    int h = tid >> 6;
    int k = tid & 63;
    const float* ae = attn + h * 48 + 32;   // a_e slice of attn[0][h]
    float sum = 0.0f;
#pragma unroll
    for (int d = 0; d < 16; ++d)
        sum += ae[d] * edge_w[(size_t)(h * 16 + d) * EDGE_FEATS + k];
    wcomb[h * EDGE_FEATS + k] = sum;
    if (k == 0) {
        float bs = 0.0f;
#pragma unroll
        for (int d = 0; d < 16; ++d) bs += ae[d] * edge_b[h * 16 + d];
        bcomb[h] = bs;
    }
}

// -------- 2) Wh = node_feats @ node_w^T + node_b via V_WMMA_F32_16X16X4_F32 --------
// One wave computes a 16x16 output tile. 8 waves/block cover all 128 out feats.
// f32 A 16x4 layout: lanes 0-15 M=lane (K=k,k+1), lanes 16-31 M=lane-16 (K=k+2,k+3).
__global__ __launch_bounds__(256) void wh_gemm(const float* __restrict__ X,
                                               const float* __restrict__ W,
                                               const float* __restrict__ bias,
                                               float* __restrict__ Wh) {
    const int lane  = threadIdx.x & 31;
    const int wave  = threadIdx.x >> 5;
    const int m0    = blockIdx.x * 16;
    const int n0    = wave * 16;
    const int l16   = lane & 15;
    const int khalf = (lane >> 4) * 2;          // 0 or 2

    const float* xrow = X + (size_t)(m0 + l16) * IN_FEATS;   // A row (M = l16)
    const float* wrow = W + (size_t)(n0 + l16) * IN_FEATS;   // B col (N = l16), B[k][n]=W[n][k]

    v8f acc = {};
#pragma unroll 4
    for (int k = 0; k < IN_FEATS; k += 4) {
        v2f a = *(const v2f*)(xrow + k + khalf);
        v2f b = *(const v2f*)(wrow + k + khalf);
        acc = __builtin_amdgcn_wmma_f32_16x16x4_f32(false, a, false, b,
                                                    (short)0, acc, false, false);
    }
    // D layout: VGPR v -> M = v + (lane>=16 ? 8 : 0), N = lane&15
    const int mbase = m0 + ((lane >> 4) << 3);
    const int ncol  = n0 + l16;
    const float bv  = bias[ncol];
#pragma unroll
    for (int v = 0; v < 8; ++v)
        Wh[(size_t)(mbase + v) * OUT_FEATS + ncol] = acc[v] + bv;
}

// -------- 3) s_src / s_dst per (node, head) --------
__global__ __launch_bounds__(256) void srcdst_scores(const float* __restrict__ Wh,
                                                     const float* __restrict__ attn,
                                                     float* __restrict__ s_src,
                                                     float* __restrict__ s_dst) {
    int gid = blockIdx.x * blockDim.x + threadIdx.x;
    if (gid >= N_NODES * NUM_HEADS) return;
    int n = gid >> 3, h = gid & 7;
    const float* w  = Wh + (size_t)n * OUT_FEATS + h * 16;
    const float* as = attn + h * 48;        // a_src
    const float* ad = as + 16;              // a_dst
    float ss = 0.0f, sd = 0.0f;
#pragma unroll
    for (int d = 0; d < 16; ++d) {
        float v = w[d];
        ss += v * as[d];
        sd += v * ad[d];
    }
    s_src[gid] = ss;
    s_dst[gid] = sd;
}

// -------- 4) edge scores: s_e via WMMA (16 edges x 8 heads, padded to 16 cols),
//            fused + s_src[src] + s_dst[dst], leaky-relu, atomicMax into m --------
__global__ __launch_bounds__(256) void edge_score(const float* __restrict__ EF,
                                                  const float* __restrict__ wcomb,
                                                  const float* __restrict__ bcomb,
                                                  const int*   __restrict__ src,
                                                  const int*   __restrict__ dst,
                                                  const float* __restrict__ s_src,
                                                  const float* __restrict__ s_dst,
                                                  float* __restrict__ escore,
                                                  float* __restrict__ mmax) {
    const int lane  = threadIdx.x & 31;
    const int wave  = threadIdx.x >> 5;
    const int e0    = (blockIdx.x * 8 + wave) * 16;
    const int l16   = lane & 15;
    const int khalf = (lane >> 4) * 2;

    const float* arow = EF + (size_t)(e0 + l16) * EDGE_FEATS;
    const float* brow = wcomb + (size_t)(l16 & 7) * EDGE_FEATS;  // cols 8..15 dup (unused)

    v8f acc = {};
#pragma unroll 4
    for (int k = 0; k < EDGE_FEATS; k += 4) {
        v2f a = *(const v2f*)(arow + k + khalf);
        v2f b = *(const v2f*)(brow + k + khalf);
        acc = __builtin_amdgcn_wmma_f32_16x16x4_f32(false, a, false, b,
                                                    (short)0, acc, false, false);
    }
    const int h = l16;
    if (h < 8) {                          // only 8 valid head columns
        const int   ebase = e0 + ((lane >> 4) << 3);
        const float bc    = bcomb[h];
#pragma unroll
        for (int v = 0; v < 8; ++v) {
            int e = ebase + v;
            int s = src[e], d = dst[e];
            float val = acc[v] + bc + s_src[s * NUM_HEADS + h] + s_dst[d * NUM_HEADS + h];
            val = val > 0.0f ? val : 0.2f * val;          // leaky_relu(0.2)
            escore[(size_t)e * NUM_HEADS + h] = val;
            atomicMaxFloat(&mmax[d * NUM_HEADS + h], val);
        }
    }
}

// -------- 5) ex = exp(e - m[dst]); z[dst] += ex (in place) --------
__global__ __launch_bounds__(256) void edge_exp(const int* __restrict__ dst,
                                                const float* __restrict__ mmax,
                                                float* __restrict__ escore,
                                                float* __restrict__ z) {
    int gid = blockIdx.x * blockDim.x + threadIdx.x;
    if (gid >= N_EDGES * NUM_HEADS) return;
    int e = gid >> 3, h = gid & 7;
    int d = dst[e];
    float ex = __expf(escore[gid] - mmax[d * NUM_HEADS + h]);
    escore[gid] = ex;
    atomicAdd(&z[d * NUM_HEADS + h], ex);
}

// -------- 6) h[dst] += (ex/z[dst]) * Wh[src] : per (edge, 4-feat quad) --------
__global__ __launch_bounds__(256) void scatter_msg(const int* __restrict__ src,
                                                   const int* __restrict__ dst,
                                                   const float* __restrict__ ex,
                                                   const float* __restrict__ z,
                                                   const float* __restrict__ Wh,
                                                   float* __restrict__ out) {
    int gid = blockIdx.x * blockDim.x + threadIdx.x;
    if (gid >= N_EDGES * 32) return;       // 32 quads of 4 feats = 128 feats
    int e  = gid >> 5;
    int q  = gid & 31;
    int f0 = q * 4;
    int h  = f0 >> 4;
    int s = src[e], d = dst[e];
    float coeff = ex[(size_t)e * NUM_HEADS + h] / z[d * NUM_HEADS + h];
    const float4 wv = *(const float4*)(Wh + (size_t)s * OUT_FEATS + f0);
    float* op = out + (size_t)d * OUT_FEATS + f0;
    atomicAdd(op + 0, coeff * wv.x);
    atomicAdd(op + 1, coeff * wv.y);
    atomicAdd(op + 2, coeff * wv.z);
    atomicAdd(op + 3, coeff * wv.w);
}

extern "C" void kernel_launch(void* const* d_in, const int* in_sizes, int n_in,
                              void* d_out, int out_size, void* d_ws, size_t ws_size,
                              hipStream_t stream) {
    const float* node_feats = (const float*)d_in[0];
    const float* edge_feats = (const float*)d_in[1];
    const int*   src        = (const int*)  d_in[2];
    const int*   dst        = (const int*)  d_in[3];
    const float* node_w     = (const float*)d_in[4];
    const float* node_b     = (const float*)d_in[5];
    const float* edge_w     = (const float*)d_in[6];
    const float* edge_b     = (const float*)d_in[7];
    const float* attn       = (const float*)d_in[8];
    float* out = (float*)d_out;

    // workspace layout (floats)
    float* ws     = (float*)d_ws;
    float* Wh     = ws;                                        // N*128
    float* s_src  = Wh    + (size_t)N_NODES * OUT_FEATS;       // N*8
    float* s_dst  = s_src + (size_t)N_NODES * NUM_HEADS;       // N*8
    float* esc    = s_dst + (size_t)N_NODES * NUM_HEADS;       // E*8
    float* mmax   = esc   + (size_t)N_EDGES * NUM_HEADS;       // N*8
    float* z      = mmax  + (size_t)N_NODES * NUM_HEADS;       // N*8
    float* wcomb  = z     + (size_t)N_NODES * NUM_HEADS;       // 8*64
    float* bcomb  = wcomb + NUM_HEADS * EDGE_FEATS;            // 8

    init_kernel<<<(N_NODES * OUT_FEATS + 255) / 256, 256, 0, stream>>>(out, mmax, z);
    prep_wcomb<<<1, 512, 0, stream>>>(edge_w, edge_b, attn, wcomb, bcomb);
    wh_gemm<<<N_NODES / 16, 256, 0, stream>>>(node_feats, node_w, node_b, Wh);
    srcdst_scores<<<(N_NODES * NUM_HEADS + 255) / 256, 256, 0, stream>>>(Wh, attn, s_src, s_dst);
    edge_score<<<N_EDGES / (16 * 8), 256, 0, stream>>>(edge_feats, wcomb, bcomb, src, dst,
                                                       s_src, s_dst, esc, mmax);
    edge_exp<<<(N_EDGES * NUM_HEADS + 255) / 256, 256, 0, stream>>>(dst, mmax, esc, z);
    scatter_msg<<<(N_EDGES * 32 + 255) / 256, 256, 0, stream>>>(src, dst, esc, z, Wh, out);
}